// bigRSNN_49615462203516
// MI455X (gfx1250) — compile-verified
//
#include <hip/hip_runtime.h>
#include <hip/hip_bf16.h>

typedef __attribute__((ext_vector_type(2))) float v2f;
typedef __attribute__((ext_vector_type(8))) float v8f;

#define T_STEPS 8192
#define IN_DIM  192
#define H_DIM   1024
#define HEADS   5
#define O_DIM   2

// ---------------------------------------------------------------------------
// Kernel 1: Icur[t][h] = b_rec[h] + sum_k W_in[h][k] * x[t-1][k]   (x[-1] = 0)
// fp32 WMMA 16x16x4: A = x-rows (16 t x 4 k), B = W_in^T tile (4 k x 16 h).
// A layout (ISA 7.12.2, 32-bit A 16x4): lane L row M=L&15; lanes 0-15 hold
// K=0,1 in the two A VGPRs, lanes 16-31 hold K=2,3. B symmetric on columns.
// C/D: VGPR r holds (row = r + 8*(lane>>4), col = lane&15).
// The t==0 row (zero input) only exists in M-tile 0, so the guard is made
// wave-uniform: fast branchless loop for tm>0, mask-multiply path for tm==0.
// ---------------------------------------------------------------------------
__global__ __launch_bounds__(256) void input_proj_wmma(
    const float* __restrict__ x, const float* __restrict__ W_in,
    const float* __restrict__ b_rec, float* __restrict__ Icur)
{
    const int lane   = threadIdx.x & 31;
    const int tile   = blockIdx.x * 8 + (threadIdx.x >> 5);  // one 16x16 tile per wave
    const int tilesN = H_DIM / 16;                           // 64
    const int tm     = tile / tilesN;                        // t-tile  0..511
    const int tn     = tile % tilesN;                        // h-tile  0..63

    const int mrow   = lane & 15;          // A row within tile
    const int khalf2 = (lane >> 4) * 2;    // 0 -> K=0,1   2 -> K=2,3
    const int hcol   = tn * 16 + (lane & 15);
    const int trow   = tm * 16 + mrow;

    const float* __restrict__ wp = W_in + (long)hcol * IN_DIM;

    v8f c = {};
    if (tm > 0) {
        // fast path: every lane's x[t-1] row is valid -> no exec-mask churn
        const float* __restrict__ xp = x + (long)(trow - 1) * IN_DIM;
#pragma unroll 4
        for (int k = 0; k < IN_DIM; k += 4) {
            const v2f a = *(const v2f*)(xp + k + khalf2);
            const v2f b = *(const v2f*)(wp + k + khalf2);
            // 8 args: (neg_a, A, neg_b, B, c_mod, C, reuse_a, reuse_b)
            c = __builtin_amdgcn_wmma_f32_16x16x4_f32(false, a, false, b,
                                                      (short)0, c, false, false);
        }
    } else {
        // tile tm==0: lane rows with trow==0 consume x[-1] == 0.
        // Clamp the address (stays in-bounds) and fold the zero in as a mul —
        // no divergent control flow, loads stay unconditional.
        const int   tclamp = (trow > 0) ? (trow - 1) : 0;
        const float amask  = (trow > 0) ? 1.0f : 0.0f;
        const float* __restrict__ xp = x + (long)tclamp * IN_DIM;
#pragma unroll 4
        for (int k = 0; k < IN_DIM; k += 4) {
            v2f a = *(const v2f*)(xp + k + khalf2);
            a.x *= amask;
            a.y *= amask;
            const v2f b = *(const v2f*)(wp + k + khalf2);
            c = __builtin_amdgcn_wmma_f32_16x16x4_f32(false, a, false, b,
                                                      (short)0, c, false, false);
        }
    }

    const float bias  = b_rec[hcol];
    const int   mbase = tm * 16 + (khalf2 >> 1) * 8;
#pragma unroll
    for (int r = 0; r < 8; ++r)
        Icur[(long)(mbase + r) * H_DIM + hcol] = c[r] + bias;
}

// ---------------------------------------------------------------------------
// Kernel 2: Vt[j][h] = V[h][j]  (so an active spike j becomes one contiguous
// 4KB row read, coalesced across the 1024 resident threads of the scan kernel)
// ---------------------------------------------------------------------------
__global__ __launch_bounds__(256) void transpose_V(
    const float* __restrict__ V, float* __restrict__ Vt)
{
    __shared__ float tile[32][33];
    const int bx = blockIdx.x % (H_DIM / 32);
    const int by = blockIdx.x / (H_DIM / 32);
    const int tx = threadIdx.x & 31;
    const int ty = threadIdx.x >> 5;                 // 0..7
    for (int i = 0; i < 32; i += 8)
        tile[ty + i][tx] = V[(long)(by * 32 + ty + i) * H_DIM + bx * 32 + tx];
    __syncthreads();
    for (int i = 0; i < 32; i += 8)
        Vt[(long)(bx * 32 + ty + i) * H_DIM + by * 32 + tx] = tile[tx][ty + i];
}

// ---------------------------------------------------------------------------
// Kernel 3: persistent sequential scan. One workgroup (32 waves), thread = h.
// Binary spikes -> compact active index list in LDS; V@spk becomes a gather
// of active Vt rows (all resident in L2: Vt is 4MB of the 192MB L2).
// ---------------------------------------------------------------------------
__global__ __launch_bounds__(1024) void rsnn_seq(
    const float* __restrict__ Icur, const float* __restrict__ Vt,
    const float* __restrict__ W_out,
    const float* __restrict__ alpha_h_p, const float* __restrict__ beta_h_p,
    const float* __restrict__ alpha_r,   const float* __restrict__ beta_r,
    float* __restrict__ out)
{
    __shared__ int   s_active[H_DIM];
    __shared__ int   s_count;
    __shared__ float s_synr[HEADS * O_DIM];
    __shared__ float s_memr[HEADS * O_DIM];

    const int   tid     = threadIdx.x;
    const int   wave    = tid >> 5;
    const int   lane    = tid & 31;
    const float alpha_h = alpha_h_p[0];
    const float beta_h  = beta_h_p[0];

    float syn_h = 0.0f, mem_h = 0.0f, spk = 0.0f;
    if (tid < HEADS * O_DIM) { s_synr[tid] = 0.0f; s_memr[tid] = 0.0f; }

    for (int t = 0; t < T_STEPS; ++t) {
        __syncthreads();                       // prev iter done reading s_active
        if (tid == 0) s_count = 0;
        __syncthreads();
        if (spk > 0.5f) {                      // compact previous-step spikes
            int p = atomicAdd(&s_count, 1);
            s_active[p] = tid;
        }
        __syncthreads();
        const int nnz = s_count;

        // hidden neuron update: cur = W_in@x[t-1] + b + sum(active cols of V)
        float cur = Icur[(long)t * H_DIM + tid];
        for (int i = 0; i < nnz; ++i)
            cur += Vt[(long)s_active[i] * H_DIM + tid];
        syn_h = alpha_h * syn_h + cur;
        mem_h = (beta_h * mem_h + syn_h) * (1.0f - spk);   // zero-reset, prev spike
        const float spk_new = (mem_h > 1.0f) ? 1.0f : 0.0f; // THRESHOLD = 1.0

        // readout heads consume the PREVIOUS spikes: one wave per (head,o)
        if (wave < HEADS * O_DIM) {
            const float* wrow = W_out + (long)wave * H_DIM;   // [head][o][:]
            float part = 0.0f;
            for (int i = lane; i < nnz; i += 32)
                part += wrow[s_active[i]];
            for (int off = 16; off > 0; off >>= 1)
                part += __shfl_down(part, off);
            if (lane == 0) {
                const int head = wave >> 1;
                const float sr = alpha_r[head] * s_synr[wave] + part;
                s_synr[wave] = sr;
                s_memr[wave] = beta_r[head] * s_memr[wave] + sr;
            }
        }
        __syncthreads();
        if (tid < O_DIM) {
            float acc = 0.0f;
#pragma unroll
            for (int hd = 0; hd < HEADS; ++hd) acc += s_memr[hd * O_DIM + tid];
            out[t * O_DIM + tid] = acc * (1.0f / HEADS);
        }
        spk = spk_new;
    }
}

extern "C" void kernel_launch(void* const* d_in, const int* in_sizes, int n_in,
                              void* d_out, int out_size, void* d_ws, size_t ws_size,
                              hipStream_t stream) {
    const float* x       = (const float*)d_in[0];   // [8192,192]
    const float* W_in    = (const float*)d_in[1];   // [1024,192]
    const float* V       = (const float*)d_in[2];   // [1024,1024]
    const float* b_rec   = (const float*)d_in[3];   // [1024]
    const float* W_out   = (const float*)d_in[4];   // [5,2,1024]
    const float* alpha_h = (const float*)d_in[5];   // scalar
    const float* beta_h  = (const float*)d_in[6];   // scalar
    const float* alpha_r = (const float*)d_in[7];   // [5]
    const float* beta_r  = (const float*)d_in[8];   // [5]
    float*       out     = (float*)d_out;           // [8192,2]

    float* Icur = (float*)d_ws;                          // 32 MB
    float* Vt   = Icur + (size_t)T_STEPS * H_DIM;        // +4 MB

    const int tiles  = (T_STEPS / 16) * (H_DIM / 16);    // 32768 wave-tiles
    input_proj_wmma<<<tiles / 8, 256, 0, stream>>>(x, W_in, b_rec, Icur);
    transpose_V<<<(H_DIM / 32) * (H_DIM / 32), 256, 0, stream>>>(V, Vt);
    rsnn_seq<<<1, 1024, 0, stream>>>(Icur, Vt, W_out,
                                     alpha_h, beta_h, alpha_r, beta_r, out);
}